// Router_33578054320453
// MI455X (gfx1250) — compile-verified
//
#include <hip/hip_runtime.h>

typedef __attribute__((ext_vector_type(2))) float v2f;
typedef __attribute__((ext_vector_type(4))) float v4f;
typedef __attribute__((ext_vector_type(8))) float v8f;

constexpr int T = 8192;   // tokens
constexpr int D = 1024;   // d_model
constexpr int E = 8;      // experts
constexpr int C = 2048;   // capacity

// ---------------------------------------------------------------------------
// Kernel 1: zero the 1.07 GiB output (the dominant cost; pure store bandwidth).
// Nontemporal b128 stores: output >> 192MB L2, so don't pollute it.
// ---------------------------------------------------------------------------
__global__ void zero_out_kernel(v4f* __restrict__ out, size_t n4) {
  size_t i = (size_t)blockIdx.x * blockDim.x + threadIdx.x;
  size_t stride = (size_t)gridDim.x * blockDim.x;
  v4f z = {0.f, 0.f, 0.f, 0.f};
  for (; i < n4; i += stride)
    __builtin_nontemporal_store(z, out + i);
}

// ---------------------------------------------------------------------------
// Kernel 2: router logits = X @ W + b via V_WMMA_F32_16X16X4_F32.
// One wave computes a 16-token x 16-col tile (cols 8..15 zero padding),
// K-loop over D in steps of 4. Exact f32 accumulation.
//
// W is staged once per block into LDS, zero-padded to 16 columns and stored
// COLUMN-major (Wlds[n][k]) so a B fragment is one aligned ds_load_b64 per
// lane — no divergent predicated global loads in the hot loop.
//
// A (16x4) layout per ISA 7.12.2: lanes 0-15 -> M=lane,    K={0,1} in v0/v1;
//                                 lanes 16-31 -> M=lane-16, K={2,3}.
// B (4x16) row-striped like C/D:  v0: K=0 (lanes 0-15) / K=2 (lanes 16-31);
//                                 v1: K=1 / K=3.
// D (16x16): VGPR r, lane l -> M = r + 8*(l>=16), N = l%16.
// ---------------------------------------------------------------------------
__global__ void __launch_bounds__(256)
router_logits_wmma(const float* __restrict__ X,
                   const float* __restrict__ W,
                   const float* __restrict__ bias,
                   float* __restrict__ logits) {
  __shared__ float Wlds[16][D];            // 64 KB, column-major padded W

  int tid  = threadIdx.x;
  int lane = tid & 31;
  int wave = (int)((blockIdx.x * blockDim.x + tid) >> 5);
  int row0 = wave * 16;                    // 16 tokens per wave
  int m    = lane & 15;                    // token row within tile (A only)
  int kh   = (lane >> 4) * 2;              // K sub-offset: 0 or 2
  int n    = lane & 15;                    // output column for B/C/D

  // stage padded W: Wlds[n][k] = (n < 8) ? W[k][n] : 0
  for (int idx = tid; idx < 16 * D; idx += 256) {
    int nn = idx & 15;
    int kk = idx >> 4;
    Wlds[nn][kk] = (nn < E) ? W[kk * E + nn] : 0.f;
  }
  __syncthreads();

  v8f c = {};
  const float* __restrict__ xrow = X + (size_t)(row0 + m) * D + kh;
  const float* __restrict__ wcol = &Wlds[n][kh];

  #pragma unroll 4
  for (int k = 0; k < D; k += 4) {
    v2f a = *(const v2f*)(xrow + k);       // global_load_b64, 8B aligned
    v2f b = *(const v2f*)(wcol + k);       // ds_load_b64,     8B aligned
    // D = A(16x4,f32) * B(4x16,f32) + C
    c = __builtin_amdgcn_wmma_f32_16x16x4_f32(
        /*neg_a=*/false, a, /*neg_b=*/false, b,
        /*c_mod=*/(short)0, c, /*reuse_a=*/false, /*reuse_b=*/false);
  }

  if (n < E) {
    float bb = bias[n];
    int mbase = row0 + ((lane >> 4) * 8);  // rows r / r+8 per lane half
    #pragma unroll
    for (int r = 0; r < 8; ++r)
      logits[(size_t)(mbase + r) * E + n] = c[r] + bb;
  }
}

// ---------------------------------------------------------------------------
// Kernel 3: single workgroup (32 wave32s) fused argmax + gate + token-order
// per-expert prefix scan + scatter. 8 chunks of 1024 tokens, persistent
// per-expert base counters in LDS. Reproduces jnp.cumsum over token index.
// ---------------------------------------------------------------------------
__global__ void __launch_bounds__(1024)
pos_scatter_kernel(const float* __restrict__ logits, float* __restrict__ out) {
  __shared__ int base[E];        // running per-expert counts across chunks
  __shared__ int wcnt[32][E];    // per-wave per-expert counts within a chunk

  int tid  = threadIdx.x;
  int lane = tid & 31;
  int wave = tid >> 5;

  if (tid < E) base[tid] = 0;
  __syncthreads();

  for (int chunk = 0; chunk < T / 1024; ++chunk) {
    int t = chunk * 1024 + tid;

    // --- softmax top-1: argmax (first index wins ties, matching top_k) ---
    float l[E];
    #pragma unroll
    for (int i = 0; i < E; ++i) l[i] = logits[(size_t)t * E + i];
    float mx = l[0]; int e = 0;
    #pragma unroll
    for (int i = 1; i < E; ++i) { if (l[i] > mx) { mx = l[i]; e = i; } }
    float s = 0.f;
    #pragma unroll
    for (int i = 0; i < E; ++i) s += __expf(l[i] - mx);
    float gate = 1.0f / s;       // softmax prob at the argmax

    // --- wave32 ballots: in-wave rank + wave totals per expert ---
    unsigned int ball[E];
    #pragma unroll
    for (int ee = 0; ee < E; ++ee)
      ball[ee] = (unsigned int)__ballot(e == ee);
    int myrank = __popc(ball[e] & ((1u << lane) - 1u));
    if (lane < E) wcnt[wave][lane] = __popc(ball[lane]);
    __syncthreads();

    // exclusive prefix over preceding waves for my expert
    int wpre = 0;
    for (int w = 0; w < wave; ++w) wpre += wcnt[w][e];
    int pos = base[e] + wpre + myrank + 1;   // 1-based, as in reference
    __syncthreads();

    // advance per-expert bases by this chunk's totals
    if (tid < E) {
      int tot = 0;
      #pragma unroll
      for (int w = 0; w < 32; ++w) tot += wcnt[w][tid];
      base[tid] += tot;
    }

    // --- scatter: keep token iff pos < C (1-based check, faithful quirk) ---
    if (pos < C) {
      size_t idx = (((size_t)t * E) + e) * C + (size_t)pos;
      out[idx] = gate;                          // dispatch_tensor
      out[(size_t)T * E * C + idx] = gate;      // combined_tensor (identical)
    }
    __syncthreads();   // wcnt/base reuse safety for next chunk
  }
}

// ---------------------------------------------------------------------------
extern "C" void kernel_launch(void* const* d_in, const int* in_sizes, int n_in,
                              void* d_out, int out_size, void* d_ws, size_t ws_size,
                              hipStream_t stream) {
  const float* X = (const float*)d_in[0];   // [T, D]
  const float* W = (const float*)d_in[1];   // [D, E]
  const float* b = (const float*)d_in[2];   // [E]
  float* out    = (float*)d_out;            // 2 * T*E*C floats
  float* logits = (float*)d_ws;             // T*E floats (256 KB scratch)

  size_t n4 = (size_t)out_size / 4;         // 67,108,864 float4 stores
  zero_out_kernel<<<4096, 256, 0, stream>>>((v4f*)out, n4);

  // 512 waves (one 16-token tile each), 8 waves per block -> 64 blocks
  router_logits_wmma<<<(T / 16) / 8, 256, 0, stream>>>(X, W, b, logits);

  pos_scatter_kernel<<<1, 1024, 0, stream>>>(logits, out);
}